// DirectionalContrastiveLoss_25348896981164
// MI455X (gfx1250) — compile-verified
//
#include <hip/hip_runtime.h>
#include <hip/hip_bf16.h>

// Problem constants (from reference)
#define NN 8
#define CC 192
#define HH 112
#define WW 112
#define HWsz (HH * WW)          // 12544
#define TILEW 16
#define NTJ (WW / TILEW)        // 7 tiles per row
#define NBCOLS 18               // j0-1 .. j0+16
#define NBROWS 3                // i-1 .. i+1
#define NBCNT (NBROWS * NBCOLS) // 54 stencil candidates
#define TEMP_INV 10.0f          // 1 / 0.1

typedef float v2f __attribute__((ext_vector_type(2)));
typedef float v8f __attribute__((ext_vector_type(8)));

__global__ void zero_acc_kernel(float* acc) { acc[0] = 0.0f; }

// Stage 1: per-pixel inverse L2 norm over channels.
__global__ void inv_norm_kernel(const float* __restrict__ f, float* __restrict__ invn) {
  int tid = blockIdx.x * blockDim.x + threadIdx.x;
  if (tid >= NN * HWsz) return;
  int n = tid / HWsz;
  int pix = tid - n * HWsz;
  const float* p = f + (size_t)n * CC * HWsz + pix;
  float ss = 0.0f;
#pragma unroll 4
  for (int c = 0; c < CC; ++c) {
    float v = p[(size_t)c * HWsz];
    ss = fmaf(v, v, ss);
  }
  invn[tid] = 1.0f / fmaxf(sqrtf(ss), 1e-12f);
}

// Stage 2: per wave, one (n, i, 16-wide j-tile). Dense 16x54 Gram via
// V_WMMA_F32_16X16X4_F32 against the full 3x18 stencil window, then select
// the 8 direction-indexed entries per pixel and accumulate the loss terms.
__global__ __launch_bounds__(256) void dcl_main_kernel(
    const float* __restrict__ f, const int* __restrict__ labels,
    const int* __restrict__ dirs, const float* __restrict__ invn,
    float* __restrict__ acc_out) {
  __shared__ float gram[8 * 16 * 64];  // 8 waves x (16 pixels x 64 candidates)

  const int lane = threadIdx.x & 31;
  const int w    = threadIdx.x >> 5;
  const int wid  = blockIdx.x * 8 + w;        // exact grid: 784*8 = 8*112*7
  const int n    = wid / (HH * NTJ);
  int rem        = wid - n * (HH * NTJ);
  const int i    = rem / NTJ;
  const int jt   = rem - i * NTJ;
  const int j0   = jt * TILEW;

  const int lo = lane & 15;   // N / M index within fragment
  const int hi = lane >> 4;   // selects K pair (A/B) or M+8 (D)

  const float* fb = f + (size_t)n * CC * HWsz;
  // A: self pixel (i, j0+lo), channels striped per ISA 16x4 f32 layout.
  const float* pA = fb + i * WW + (j0 + lo);
  // B tiles: candidate nb = t*16 + lo; clamp nb>=54 to a safe duplicate
  // (garbage Gram columns 54..63 are never selected) -> no divergence.
  const float* pB[4];
#pragma unroll
  for (int t = 0; t < 4; ++t) {
    int nb = t * 16 + lo;
    if (nb > NBCNT - 1) nb = NBCNT - 1;
    int r   = nb / NBCOLS;
    int jj  = nb - r * NBCOLS;
    int row = i - 1 + r;  row = row < 0 ? 0 : (row > HH - 1 ? HH - 1 : row);
    int col = j0 - 1 + jj; col = col < 0 ? 0 : (col > WW - 1 ? WW - 1 : col);
    pB[t] = fb + row * WW + col;
  }

  v8f acc0 = {}, acc1 = {}, acc2 = {}, acc3 = {};
  const int cb = 2 * hi;  // lanes 0-15: K=0,1 ; lanes 16-31: K=2,3
  for (int k = 0; k < CC / 4; ++k) {
    const size_t c0 = (size_t)(4 * k + cb) * HWsz;
    const size_t c1 = c0 + HWsz;
    v2f av;  av.x  = pA[c0];    av.y  = pA[c1];
    v2f b0v; b0v.x = pB[0][c0]; b0v.y = pB[0][c1];
    v2f b1v; b1v.x = pB[1][c0]; b1v.y = pB[1][c1];
    v2f b2v; b2v.x = pB[2][c0]; b2v.y = pB[2][c1];
    v2f b3v; b3v.x = pB[3][c0]; b3v.y = pB[3][c1];
    acc0 = __builtin_amdgcn_wmma_f32_16x16x4_f32(false, av, false, b0v, (short)0, acc0, false, false);
    acc1 = __builtin_amdgcn_wmma_f32_16x16x4_f32(false, av, false, b1v, (short)0, acc1, false, false);
    acc2 = __builtin_amdgcn_wmma_f32_16x16x4_f32(false, av, false, b2v, (short)0, acc2, false, false);
    acc3 = __builtin_amdgcn_wmma_f32_16x16x4_f32(false, av, false, b3v, (short)0, acc3, false, false);
  }

  // Spill Gram to LDS: D layout is M = e + 8*hi, N = lo.
  float* g = &gram[w * 16 * 64];
#pragma unroll
  for (int e = 0; e < 8; ++e) {
    int prow = e + 8 * hi;
    g[prow * 64 + 0 * 16 + lo] = acc0[e];
    g[prow * 64 + 1 * 16 + lo] = acc1[e];
    g[prow * 64 + 2 * 16 + lo] = acc2[e];
    g[prow * 64 + 3 * 16 + lo] = acc3[e];
  }
  __syncthreads();

  // Phase B: 2 lanes per pixel, 4 m's each.
  const int p     = lane >> 1;
  const int mbase = (lane & 1) * 4;
  const int jp    = j0 + p;
  const float inv_a = invn[n * HWsz + i * WW + jp];

  float logit[4], maskf[4];
  float esum = 0.0f;
#pragma unroll
  for (int mi = 0; mi < 4; ++mi) {
    int m  = mbase + mi;
    int d0 = dirs[((m * 2 + 0) * HH + i) * WW + jp];
    int d1 = dirs[((m * 2 + 1) * HH + i) * WW + jp];
    int row = i + d0, col = jp + d1;
    int nb  = (d0 + 1) * NBCOLS + (p + d1 + 1);
    float dot   = g[p * 64 + nb];
    float inv_b = invn[n * HWsz + row * WW + col];
    float lg    = dot * inv_a * inv_b * TEMP_INV;
    int labm = labels[(m * HH + i) * WW + jp];
    int labg = labels[(n * HH + row) * WW + col];
    float mk = (labm == labg) ? 1.0f : 0.0f;
    logit[mi] = lg;
    maskf[mi] = mk;
    esum += mk * expf(lg);
  }
  float denom = esum + __shfl_xor(esum, 1, 32);  // join the two lanes of a pixel
  float ld = logf(denom + 1e-6f);
  float contrib = 0.0f;
#pragma unroll
  for (int mi = 0; mi < 4; ++mi) {
    float lt = (maskf[mi] > 0.0f) ? logit[mi] : -__builtin_inff();
    contrib += ld - lt;
  }
  // wave32 reduction, one atomic per wave
#pragma unroll
  for (int off = 16; off >= 1; off >>= 1)
    contrib += __shfl_xor(contrib, off, 32);
  if (lane == 0) atomicAdd(acc_out, contrib);
}

__global__ void finalize_kernel(const float* __restrict__ acc, float* __restrict__ out) {
  out[0] = acc[0] * (1.0f / (float)(NN * 8 * HH * WW));  // /802816
}

extern "C" void kernel_launch(void* const* d_in, const int* in_sizes, int n_in,
                              void* d_out, int out_size, void* d_ws, size_t ws_size,
                              hipStream_t stream) {
  const float* features = (const float*)d_in[0];
  const int*   labels   = (const int*)d_in[1];
  const int*   dirs     = (const int*)d_in[2];
  float* acc  = (float*)d_ws;       // [0]: loss accumulator
  float* invn = acc + 64;           // 8*112*112 floats of inverse norms

  zero_acc_kernel<<<1, 1, 0, stream>>>(acc);
  inv_norm_kernel<<<(NN * HWsz + 255) / 256, 256, 0, stream>>>(features, invn);
  dcl_main_kernel<<<NN * HH * NTJ / 8, 256, 0, stream>>>(features, labels, dirs, invn, acc);
  finalize_kernel<<<1, 1, 0, stream>>>(acc, (float*)d_out);
}